// PostProcess_18811956757112
// MI455X (gfx1250) — compile-verified
//
#include <hip/hip_runtime.h>
#include <hip/hip_bf16.h>

// Problem constants (from reference)
#define Bn 16
#define Nn 10647
#define Cn 80
#define Kn 2048
#define Fn 8192             // filtered-candidate capacity per batch (pow2 for bitonic)
#define SCORE_THR 0.5f
#define IOU_THR   0.5f

typedef unsigned int       u32;
typedef unsigned long long u64;

typedef unsigned int u32x4 __attribute__((ext_vector_type(4)));
typedef int          i32x8 __attribute__((ext_vector_type(8)));
typedef int          i32x4 __attribute__((ext_vector_type(4)));

// ---------------------------------------------------------------------------
// TDM: 1-D DMA of `bytes` (multiple of 8, /8 <= 65535) from global -> LDS.
// D# layout per CDNA5 ISA:
//   group0: [1:0]=count=1, [63:32]=lds_addr, [120:64]=global_addr, [127:126]=type=2
//   group1: [17:16]=data_size (3 => 8B), [79:48]=tensor_dim0, [111:80]=tensor_dim1,
//           [127:112]=tile_dim0, [207:160]=tensor_dim0_stride
// Issued by one wave; completion tracked with TENSORcnt.
// ---------------------------------------------------------------------------
__device__ __forceinline__ void tdm_load_1d_to_lds(u32 lds_off, const void* gptr, u32 bytes) {
  u64 ga    = (u64)(size_t)gptr;
  u32 units = bytes >> 3;                       // 8-byte units
  u32x4 g0 = { 1u,                              // count=1, user mode
               lds_off,                         // lds_addr (bytes)
               (u32)ga,                         // global_addr[31:0]
               (u32)((ga >> 32) & 0x01FFFFFFu) | 0x80000000u }; // addr[56:32] | type=2
  i32x8 g1 = { (int)(3u << 16),                 // data_size = 3 (8B); wg_mask=0
               (int)((units & 0xFFFFu) << 16),  // tensor_dim0[15:0] in bits 63:48
               (int)((units >> 16) | (1u << 16)), // tensor_dim0[31:16] | tensor_dim1=1
               (int)((units & 0xFFFFu) << 16),  // tile_dim0 in bits 127:112
               0,                               // tile_dim1=0 (1-D), tile_dim2=0
               (int)units,                      // tensor_dim0_stride[31:0]
               0, 0 };
  i32x4 z4 = { 0, 0, 0, 0 };
#if defined(__clang_major__) && (__clang_major__ >= 23)
  i32x8 z8 = { 0, 0, 0, 0, 0, 0, 0, 0 };
  __builtin_amdgcn_tensor_load_to_lds(g0, g1, z4, z4, z8, 0);
#else
  __builtin_amdgcn_tensor_load_to_lds(g0, g1, z4, z4, 0);
#endif
}

__device__ __forceinline__ bool iou_over(const float4& a, float areaA,
                                         const float4& b, float areaB) {
  float ix1 = fmaxf(a.x, b.x);
  float iy1 = fmaxf(a.y, b.y);
  float ix2 = fminf(a.z, b.z);
  float iy2 = fminf(a.w, b.w);
  float inter = fmaxf(ix2 - ix1, 0.0f) * fmaxf(iy2 - iy1, 0.0f);
  float uni = areaA + areaB - inter;
  return inter > IOU_THR * fmaxf(uni, 1e-9f);   // iou > IOU_THR
}

// ---------------------------------------------------------------------------
// Kernel 1: streaming score pass + threshold compaction.
// Slots with score <= 0.5 always produce zero output (keep starts false and
// out = value*keep), and descending sort keeps the >thr prefix order intact,
// so only >thr candidates need sorting. ~5.3K of 10647 pass per batch here;
// Fn=8192 capacity is a huge margin. key = (score_bits<<32) | ~idx gives
// lax.top_k semantics (descending, ties -> lowest index) under one u64 sort.
// Memory-bound: 58 MB read once (~2.5 us at 23.3 TB/s).
// ---------------------------------------------------------------------------
__global__ __launch_bounds__(256) void score_key_kernel(const float* __restrict__ y,
                                                        u64* __restrict__ fkeys,
                                                        int* __restrict__ cnt) {
  int t = blockIdx.x * 256 + threadIdx.x;          // 0 .. Bn*Nn-1
  if (t >= Bn * Nn) return;
  int b = t / Nn;
  int i = t - b * Nn;
  const float* p = y + (size_t)t * (5 + Cn);
  __builtin_prefetch(p, 0, 0);                     // global_prefetch_b8
  float obj = p[4];
  float m = p[5];
#pragma unroll 4
  for (int c = 6; c < 5 + Cn; ++c) m = fmaxf(m, p[c]);
  float s = obj * m;                               // s >= 0: float bits monotone
  if (s > SCORE_THR) {
    u64 key = ((u64)__float_as_uint(s) << 32) | (u32)(~(u32)i);
    int pos = atomicAdd(&cnt[b], 1);
    if (pos < Fn) fkeys[(size_t)b * Fn + pos] = key;
  }
}

// ---------------------------------------------------------------------------
// Kernel 2: per-batch bitonic sort of 8192 64-bit keys entirely in LDS
// (64 KB tile, TDM-DMA'd in; WGP has 320 KB). 91 compare-exchange passes,
// 1024 threads/batch. Then gather top-K boxes -> corners + scores.
// ---------------------------------------------------------------------------
__global__ __launch_bounds__(1024) void sort_gather_kernel(const u64* __restrict__ fkeys,
                                                           const float* __restrict__ y,
                                                           float4* __restrict__ corners,
                                                           float* __restrict__ scores) {
  extern __shared__ u64 sKeys[];                   // Fn * 8 = 65536 bytes
  const int b = blockIdx.x, tid = threadIdx.x;

  if (tid == 0) {
    tdm_load_1d_to_lds((u32)(size_t)&sKeys[0], fkeys + (size_t)b * Fn, Fn * 8);
  }
  __builtin_amdgcn_s_wait_tensorcnt(0);            // issuing wave waits; no-op for others
  __syncthreads();

  for (int k = 2; k <= Fn; k <<= 1) {
    for (int j = k >> 1; j > 0; j >>= 1) {
      for (int i = tid; i < Fn; i += 1024) {
        int ixj = i ^ j;
        if (ixj > i) {
          u64 a = sKeys[i], c = sKeys[ixj];
          bool desc = ((i & k) == 0);              // overall descending sort
          bool sw = desc ? (a < c) : (a > c);
          if (sw) { sKeys[i] = c; sKeys[ixj] = a; }
        }
      }
      __syncthreads();
    }
  }

  // Gather top-K: xywh -> corners, write to workspace (lands in the 192MB L2)
  for (int q = tid; q < Kn; q += 1024) {
    u64 key = sKeys[q];
    float4 cr = make_float4(0.0f, 0.0f, 0.0f, 0.0f);
    float s = 0.0f;
    if (key != 0ull) {                             // padding guard (no OOB gather)
      s = __uint_as_float((u32)(key >> 32));
      u32 n = ~(u32)key;
      const float* p = y + ((size_t)b * Nn + n) * (5 + Cn);
      float x = p[0], yc = p[1], w = p[2], h = p[3];
      cr = make_float4(x - 0.5f * w, yc - 0.5f * h, x + 0.5f * w, yc + 0.5f * h);
    }
    corners[(size_t)b * Kn + q] = cr;
    scores[(size_t)b * Kn + q] = s;
  }
}

// ---------------------------------------------------------------------------
// Kernel 3: blocked greedy NMS + masked output. One 1024-thread block/batch.
// The 32 KB corner tile is TDM-DMA'd into LDS. Greedy order is honored in
// tiles of 32: wave 0 resolves intra-tile suppression wave-synchronously
// (ballot loop, zero barriers), then all threads apply the tile's kept boxes
// to every j beyond the tile. 2 barriers per tile -> 128 total (vs 2048).
// Matches reference: only kept boxes suppress; suppression of j>i with
// IoU>thr is unconditional on j's current state.
// ---------------------------------------------------------------------------
__global__ __launch_bounds__(1024) void nms_out_kernel(const float4* __restrict__ corners,
                                                       const float* __restrict__ scores,
                                                       float* __restrict__ out) {
  __shared__ float4        sBox[Kn];               // 32 KB
  __shared__ float         sArea[Kn];              // 8 KB
  __shared__ unsigned char sKeep[Kn];              // 2 KB
  __shared__ u32           sTileKept;
  const int b = blockIdx.x, tid = threadIdx.x;
  const int lane = tid & 31, wid = tid >> 5;

  if (tid == 0) {
    tdm_load_1d_to_lds((u32)(size_t)&sBox[0], corners + (size_t)b * Kn, Kn * 16);
  }
  __builtin_amdgcn_s_wait_tensorcnt(0);
  __syncthreads();

  for (int i = tid; i < Kn; i += 1024) {
    float4 bx = sBox[i];
    sArea[i] = fmaxf(bx.z - bx.x, 0.0f) * fmaxf(bx.w - bx.y, 0.0f);
    sKeep[i] = (scores[(size_t)b * Kn + i] > SCORE_THR) ? 1 : 0;
  }
  __syncthreads();

  for (int t0 = 0; t0 < Kn; t0 += 32) {
    // --- step 1: wave 0 resolves greedy suppression inside the tile -------
    if (wid == 0) {
      float4 bm = sBox[t0 + lane];
      float  am = sArea[t0 + lane];
      bool myAlive = (sKeep[t0 + lane] != 0);
      u32 sup = 0;                                 // bit s: tile box s suppresses me
      for (int s = 0; s < 32; ++s) {
        if (s < lane && iou_over(sBox[t0 + s], sArea[t0 + s], bm, am)) sup |= 1u << s;
      }
      u32 kept = 0;
      for (int s = 0; s < 32; ++s) {               // sequential greedy, 32 ballots
        u32 aliveMask = (u32)__ballot(myAlive);
        if ((aliveMask >> s) & 1u) {
          kept |= 1u << s;
          if (lane > s && ((sup >> s) & 1u)) myAlive = false;
        }
      }
      sKeep[t0 + lane] = myAlive ? 1 : 0;
      if (lane == 0) sTileKept = kept;
    }
    __syncthreads();

    // --- step 2: apply tile's kept boxes to all j beyond the tile ---------
    const u32 kept = sTileKept;
    if (kept) {
      for (int j = t0 + 32 + tid; j < Kn; j += 1024) {
        if (sKeep[j]) {
          float4 bj = sBox[j];
          float  aj = sArea[j];
          u32 m = kept;
          bool dead = false;
          while (m) {
            int s = __ffs(m) - 1;
            m &= m - 1;
            if (iou_over(sBox[t0 + s], sArea[t0 + s], bj, aj)) { dead = true; break; }
          }
          if (dead) sKeep[j] = 0;
        }
      }
    }
    __syncthreads();
  }

  for (int q = tid; q < Kn; q += 1024) {
    float kf = sKeep[q] ? 1.0f : 0.0f;
    float4 bx = sBox[q];
    float s = scores[(size_t)b * Kn + q];
    float* o = out + ((size_t)b * Kn + q) * 5;
    o[0] = bx.x * kf;
    o[1] = bx.y * kf;
    o[2] = bx.z * kf;
    o[3] = bx.w * kf;
    o[4] = s * kf;
  }
}

// ---------------------------------------------------------------------------
extern "C" void kernel_launch(void* const* d_in, const int* in_sizes, int n_in,
                              void* d_out, int out_size, void* d_ws, size_t ws_size,
                              hipStream_t stream) {
  const float* y = (const float*)d_in[0];          // (B, N, 85) float32
  float* out = (float*)d_out;                      // (B, K, 5)  float32

  char* ws = (char*)d_ws;
  u64*    fkeys   = (u64*)ws;                                          // 1 MB
  float4* corners = (float4*)(ws + (size_t)Bn * Fn * sizeof(u64));     // 512 KB
  float*  scores  = (float*)(ws + (size_t)Bn * Fn * sizeof(u64)
                                + (size_t)Bn * Kn * sizeof(float4));   // 128 KB
  int*    cnt     = (int*)(ws + (size_t)Bn * Fn * sizeof(u64)
                              + (size_t)Bn * Kn * sizeof(float4)
                              + (size_t)Bn * Kn * sizeof(float));      // 64 B

  // workspace is not re-poisoned between replays: zero the parts we append to
  (void)hipMemsetAsync(fkeys, 0, (size_t)Bn * Fn * sizeof(u64), stream);
  (void)hipMemsetAsync(cnt, 0, Bn * sizeof(int), stream);

  // 64 KB dynamic LDS for the bitonic sort (gfx1250 WGP has 320 KB)
  (void)hipFuncSetAttribute((const void*)sort_gather_kernel,
                            hipFuncAttributeMaxDynamicSharedMemorySize,
                            Fn * (int)sizeof(u64));

  score_key_kernel<<<(Bn * Nn + 255) / 256, 256, 0, stream>>>(y, fkeys, cnt);
  sort_gather_kernel<<<Bn, 1024, Fn * sizeof(u64), stream>>>(fkeys, y, corners, scores);
  nms_out_kernel<<<Bn, 1024, 0, stream>>>(corners, scores, out);
}